// Graph_to_CNN_56538949484911
// MI455X (gfx1250) — compile-verified
//
#include <hip/hip_runtime.h>

// Problem constants (match reference: B,N,C,H,W = 32,8192,64,64,64)
constexpr int B_ = 32;
constexpr int N_ = 8192;
constexpr int C_ = 64;
constexpr int H_ = 64;
constexpr int W_ = 64;
constexpr int CELLS = H_ * W_;                 // 4096 cells per batch
constexpr int SUMS_ELEMS = B_ * CELLS * C_;    // 8,388,608 floats (32 MiB), HWC layout
constexpr int CNT_ELEMS  = B_ * CELLS;         // 131,072 floats

typedef __attribute__((ext_vector_type(2))) float v2f;
typedef __attribute__((ext_vector_type(8))) float v8f;

// ---------------------------------------------------------------------------
// Kernel 1: zero the workspace (sums + counts)
// ---------------------------------------------------------------------------
__global__ void zero_ws_kernel(float* __restrict__ ws, int n) {
    int i = blockIdx.x * blockDim.x + threadIdx.x;
    if (i < n) ws[i] = 0.0f;
}

// ---------------------------------------------------------------------------
// Kernel 2: scatter-add features into HWC sum buffer (coalesced f32 atomics)
// 16 threads per node; each thread handles 4 consecutive channels (float4).
// ---------------------------------------------------------------------------
__global__ void scatter_add_kernel(const float* __restrict__ feat,
                                   const int*   __restrict__ locs,
                                   float* __restrict__ sums,
                                   float* __restrict__ counts) {
    const int gid  = blockIdx.x * 256 + threadIdx.x;
    const int node = gid >> 4;     // B*N nodes total (exact grid, no guard needed)
    const int q    = gid & 15;     // which float4 of the 64-channel row

    const int y = locs[2 * node + 0];
    const int x = locs[2 * node + 1];
    const int b = node >> 13;      // node / N_ (N_ = 8192)
    const int cell = b * CELLS + y * W_ + x;

    const float4 f = reinterpret_cast<const float4*>(feat)[node * 16 + q];

    float* dst = sums + (size_t)cell * C_ + q * 4;   // 16 B contiguous per lane
    atomicAdd(dst + 0, f.x);
    atomicAdd(dst + 1, f.y);
    atomicAdd(dst + 2, f.z);
    atomicAdd(dst + 3, f.w);

    if (q == 0) atomicAdd(&counts[cell], 1.0f);
}

// ---------------------------------------------------------------------------
// Kernel 3: fused normalize + HWC->CHW transpose via V_WMMA_F32_16X16X4_F32.
//
// Per wave: one 16-cell x 16-channel tile.  D = S^T x diag(1/count):
//   A[m][k] = sums[cell k][channel m]  (m = channel, k = cell)   -> 16x16 via 4 K-chunks
//   B[k][n] = (k==n) ? 1/max(count[n],1) : 0                     (diagonal, in regs)
//   D[m][n] = sums[cell n][channel m] / max(count[n],1)
// D layout (M over 8 VGPRs, N over lanes) == CHW order -> contiguous stores.
// EXEC is all-ones: exact grid, zero divergence around the WMMAs.
// ---------------------------------------------------------------------------
__global__ void wmma_norm_transpose_kernel(const float* __restrict__ sums,
                                           const float* __restrict__ counts,
                                           float* __restrict__ out) {
    const int lane = threadIdx.x & 31;
    const int wave = threadIdx.x >> 5;
    const int t    = blockIdx.x * 8 + wave;      // tile id, 32768 total

    const int cb     = t & 3;                    // channel block (16 channels)
    const int cell16 = (t >> 2) & 255;           // cell block (16 cells)
    const int b      = t >> 10;                  // batch

    const int lm = lane & 15;
    const int hs = lane >> 4;                    // half-wave select

    const int cellbase = b * CELLS + cell16 * 16;
    const float* S = sums + (size_t)cellbase * C_ + cb * 16;

    // reciprocal count for cell `lm` of this tile (ref: 1/max(count,1))
    const float cnt = counts[cellbase + lm];
    const float r = 1.0f / fmaxf(cnt, 1.0f);

    v8f acc = {};
#pragma unroll
    for (int kc = 0; kc < 4; ++kc) {
        const int row0 = 4 * kc + 2 * hs;        // K index held by this half-wave, VGPR0
        v2f a;                                   // A 16x4 chunk: lane holds M=lm, K=row0/row0+1
        a.x = S[(size_t)row0 * C_ + lm];         // lanes 0-15: 64B contiguous
        a.y = S[(size_t)(row0 + 1) * C_ + lm];
        v2f bm;                                  // B 4x16 chunk: diagonal of reciprocals
        bm.x = (lm == row0)     ? r : 0.0f;
        bm.y = (lm == row0 + 1) ? r : 0.0f;
        // 8 args: (neg_a, A, neg_b, B, c_mod, C, reuse_a, reuse_b)
        acc = __builtin_amdgcn_wmma_f32_16x16x4_f32(
            false, a, false, bm, (short)0, acc, false, false);
    }

    // D: VGPR j -> channel cb*16 + j + 8*hs, cell = cell16*16 + lm  (CHW order)
    float* obase = out + ((size_t)b * C_ + cb * 16 + 8 * hs) * CELLS
                       + cell16 * 16 + lm;
#pragma unroll
    for (int j = 0; j < 8; ++j)
        obase[(size_t)j * CELLS] = acc[j];
}

// ---------------------------------------------------------------------------
extern "C" void kernel_launch(void* const* d_in, const int* in_sizes, int n_in,
                              void* d_out, int out_size, void* d_ws, size_t ws_size,
                              hipStream_t stream) {
    const float* feat = (const float*)d_in[0];   // [B, N, C] f32
    const int*   locs = (const int*)d_in[1];     // [B, N, 2] i32

    float* sums   = (float*)d_ws;                // [B, H*W, C] f32 (HWC)
    float* counts = sums + SUMS_ELEMS;           // [B, H*W]   f32
    float* out    = (float*)d_out;               // [B, C, H, W] f32

    // 1) zero sums + counts
    const int ztot = SUMS_ELEMS + CNT_ELEMS;
    zero_ws_kernel<<<(ztot + 255) / 256, 256, 0, stream>>>(sums, ztot);

    // 2) scatter-add (B*N*16 threads, exact multiple of 256)
    scatter_add_kernel<<<(B_ * N_ * 16) / 256, 256, 0, stream>>>(
        feat, locs, sums, counts);

    // 3) WMMA normalize + transpose: 32768 tiles, 8 waves/block -> 4096 blocks
    wmma_norm_transpose_kernel<<<(B_ * (CELLS / 16) * (C_ / 16)) / 8, 256, 0, stream>>>(
        sums, counts, out);
}